// GlimpseExtractor_79439715106821
// MI455X (gfx1250) — compile-verified
//
#include <hip/hip_runtime.h>
#include <math.h>

// Problem constants (from reference)
constexpr int B_  = 16;
constexpr int N_  = 16384;
constexpr int D_  = 1024;
constexpr int DV_ = 512;
constexpr int TOPK = 16;
constexpr int SL  = 32;              // n-slices for streaming reduction
constexpr int ROWS = N_ / SL;        // 512 rows per slice

// Workspace layout (float offsets)
constexpr size_t OFF_PA     = 0;                         // partial sum(alpha*F): B*SL*D
constexpr size_t OFF_PT     = OFF_PA + (size_t)B_*SL*D_; // partial sum(F):       B*SL*D
constexpr size_t OFF_TIDX   = OFF_PT + (size_t)B_*SL*D_; // topk indices (int)    B*16
constexpr size_t OFF_TW     = OFF_TIDX + B_*TOPK;        // topk weights          B*16
constexpr size_t OFF_ATOT   = OFF_TW + B_*TOPK;          // sum(alpha) per b      B
constexpr size_t OFF_WSUM   = OFF_ATOT + B_;             // sum(w) per b          B
constexpr size_t OFF_SCALE  = OFF_WSUM + B_;             // per-row bias scale    48
constexpr size_t OFF_CONCAT = OFF_SCALE + 48;            // (B, 3*DV)             16*1536
constexpr size_t OFF_H      = OFF_CONCAT + (size_t)B_*3*DV_; // (B, 2*DV->1024)   16*1024
constexpr size_t OFF_X      = OFF_H + (size_t)B_*1024;   // (48, D)               48*1024
// total ~1.14M floats (~4.6 MB) of d_ws

typedef float v2f __attribute__((ext_vector_type(2)));
typedef float v8f __attribute__((ext_vector_type(8)));

// ---------------------------------------------------------------------------
// 1) Per-batch top-16 of alpha + sum(alpha). One block per batch, 256 threads,
//    each thread owns 64 strided values in registers; deterministic reductions.
// ---------------------------------------------------------------------------
__global__ void topk_kernel(const float* __restrict__ alpha, float* __restrict__ ws) {
    const int b = blockIdx.x, t = threadIdx.x;
    __shared__ float rv[256];
    __shared__ int   ri[256];
    __shared__ float sel_v[TOPK];
    __shared__ int   sel_i[TOPK];
    __shared__ float sS;

    float vals[64];
    const float* ap = alpha + (size_t)b * N_;
#pragma unroll
    for (int i = 0; i < 64; ++i) vals[i] = ap[t + i * 256];

    // sum(alpha)
    float ts = 0.f;
#pragma unroll
    for (int i = 0; i < 64; ++i) ts += vals[i];
    rv[t] = ts; __syncthreads();
    for (int off = 128; off > 0; off >>= 1) {
        if (t < off) rv[t] += rv[t + off];
        __syncthreads();
    }
    const float aTot = rv[0];
    __syncthreads();

    unsigned long long consumed = 0ull;
    for (int r = 0; r < TOPK; ++r) {
        float best = -1e30f; int bi = 1 << 30;
#pragma unroll
        for (int i = 0; i < 64; ++i) {
            const int n = t + i * 256;
            const bool avail = ((consumed >> i) & 1ull) == 0ull;
            const float v = vals[i];
            if (avail && (v > best || (v == best && n < bi))) { best = v; bi = n; }
        }
        rv[t] = best; ri[t] = bi; __syncthreads();
        for (int off = 128; off > 0; off >>= 1) {
            if (t < off) {
                const float v2 = rv[t + off]; const int i2 = ri[t + off];
                if (v2 > rv[t] || (v2 == rv[t] && i2 < ri[t])) { rv[t] = v2; ri[t] = i2; }
            }
            __syncthreads();
        }
        const int win = ri[0]; const float wval = rv[0];
        if ((win & 255) == t) consumed |= (1ull << (win >> 8));
        if (t == 0) { sel_v[r] = wval; sel_i[r] = win; }
        __syncthreads();
    }

    if (t == 0) {
        float s = 0.f;
        for (int k = 0; k < TOPK; ++k) s += sel_v[k];
        sS = s;
        ws[OFF_ATOT + b] = aTot;
        ws[OFF_WSUM + b] = s / (s + 1e-8f);
    }
    __syncthreads();
    if (t < TOPK) {
        int* tix = (int*)(ws + OFF_TIDX);
        tix[b * TOPK + t] = sel_i[t];
        ws[OFF_TW + b * TOPK + t] = sel_v[t] / (sS + 1e-8f);
    }
}

// ---------------------------------------------------------------------------
// 2) Streaming reduction over F (the 1 GB pass): per (slice, batch) block,
//    deterministic partial sums  Σ F  and  Σ alpha*F  over 512 rows.
//    256 threads * float4 covers the full D=1024 row; coalesced b128 loads.
// ---------------------------------------------------------------------------
__global__ void stream_reduce_kernel(const float* __restrict__ F,
                                     const float* __restrict__ alpha,
                                     float* __restrict__ ws) {
    const int s = blockIdx.x, b = blockIdx.y, t = threadIdx.x;
    __shared__ float sa[ROWS];
    const int n0 = s * ROWS;
    sa[t]       = alpha[(size_t)b * N_ + n0 + t];
    sa[t + 256] = alpha[(size_t)b * N_ + n0 + t + 256];
    __syncthreads();

    const float* fp = F + ((size_t)b * N_ + n0) * D_ + t * 4;
    float t0 = 0, t1 = 0, t2 = 0, t3 = 0;
    float a0 = 0, a1 = 0, a2 = 0, a3 = 0;
#pragma unroll 4
    for (int n = 0; n < ROWS; ++n) {
        const float4 f = *(const float4*)(fp + (size_t)n * D_);
        const float a = sa[n];
        t0 += f.x; t1 += f.y; t2 += f.z; t3 += f.w;
        a0 += a * f.x; a1 += a * f.y; a2 += a * f.z; a3 += a * f.w;
    }
    const size_t o = ((size_t)(b * SL + s)) * D_ + t * 4;
    *(float4*)(ws + OFF_PA + o) = make_float4(a0, a1, a2, a3);
    *(float4*)(ws + OFF_PT + o) = make_float4(t0, t1, t2, t3);
}

// ---------------------------------------------------------------------------
// 3) Build X (48 x 1024): rows {focus, broad, periph} per batch + bias scales.
// ---------------------------------------------------------------------------
__global__ void build_x_kernel(const float* __restrict__ F, float* __restrict__ ws) {
    const int b = blockIdx.x, t = threadIdx.x;
    __shared__ int   sidx[TOPK];
    __shared__ float swt[TOPK];
    if (t < TOPK) {
        sidx[t] = ((const int*)(ws + OFF_TIDX))[b * TOPK + t];
        swt[t]  = ws[OFF_TW + b * TOPK + t];
    }
    __syncthreads();

    const int d = t * 4;
    float br0 = 0, br1 = 0, br2 = 0, br3 = 0;   // sum alpha*F
    float tt0 = 0, tt1 = 0, tt2 = 0, tt3 = 0;   // sum F
    for (int s = 0; s < SL; ++s) {
        const size_t o = ((size_t)(b * SL + s)) * D_ + d;
        const float4 pa = *(const float4*)(ws + OFF_PA + o);
        const float4 pt = *(const float4*)(ws + OFF_PT + o);
        br0 += pa.x; br1 += pa.y; br2 += pa.z; br3 += pa.w;
        tt0 += pt.x; tt1 += pt.y; tt2 += pt.z; tt3 += pt.w;
    }
    float fo0 = 0, fo1 = 0, fo2 = 0, fo3 = 0;   // weighted top-k
    float un0 = 0, un1 = 0, un2 = 0, un3 = 0;   // unweighted top-k
    for (int k = 0; k < TOPK; ++k) {
        const float4 f = *(const float4*)(F + ((size_t)b * N_ + sidx[k]) * D_ + d);
        const float w = swt[k];
        fo0 += w * f.x; fo1 += w * f.y; fo2 += w * f.z; fo3 += w * f.w;
        un0 += f.x; un1 += f.y; un2 += f.z; un3 += f.w;
    }
    const float inv = 1.0f / (float)(N_ - TOPK);
    float* X = ws + OFF_X;
    *(float4*)(X + (size_t)(3 * b + 0) * D_ + d) = make_float4(fo0, fo1, fo2, fo3);
    *(float4*)(X + (size_t)(3 * b + 1) * D_ + d) = make_float4(br0, br1, br2, br3);
    *(float4*)(X + (size_t)(3 * b + 2) * D_ + d) =
        make_float4((tt0 - un0) * inv, (tt1 - un1) * inv, (tt2 - un2) * inv, (tt3 - un3) * inv);
    if (t == 0) {
        ws[OFF_SCALE + 3 * b + 0] = ws[OFF_WSUM + b];
        ws[OFF_SCALE + 3 * b + 1] = ws[OFF_ATOT + b];
        ws[OFF_SCALE + 3 * b + 2] = 1.0f;
    }
}

// ---------------------------------------------------------------------------
// 4) Y(48,512) = X(48,1024) @ Wv[step]^T, bias scaled per row, scattered
//    straight into the concat buffer layout. One 16x16 tile per wave,
//    V_WMMA_F32_16X16X4_F32 over K.  grid 24 x 128 threads = 96 waves = tiles.
// ---------------------------------------------------------------------------
__global__ void wmma_gemm_v_kernel(const float* __restrict__ Wv_w,
                                   const float* __restrict__ Wv_b,
                                   const int* __restrict__ stepPtr,
                                   float* __restrict__ ws) {
    const int step = *stepPtr;
    const float* W  = Wv_w + (size_t)step * DV_ * D_;
    const float* bv = Wv_b + (size_t)step * DV_;
    const float* X  = ws + OFF_X;
    const float* sc = ws + OFF_SCALE;
    float* C = ws + OFF_CONCAT;

    const int wv = threadIdx.x >> 5, lane = threadIdx.x & 31;
    const int tile = blockIdx.x * 4 + wv;          // 0..95
    const int mt = tile % 3, nt = tile / 3;
    const int r0 = mt * 16, v0 = nt * 16;
    const int m = lane & 15, half = lane >> 4;

    const float* xp = X + (size_t)(r0 + m) * D_ + 2 * half;
    const float* wp = W + (size_t)(v0 + m) * D_ + 2 * half;   // B col index == lane&15
    v8f acc = {};
    for (int k = 0; k < D_; k += 4) {
        const v2f a  = *(const v2f*)(xp + k);
        const v2f bb = *(const v2f*)(wp + k);
        acc = __builtin_amdgcn_wmma_f32_16x16x4_f32(false, a, false, bb, (short)0, acc, false, false);
    }
    const int col = v0 + m;
    const float bcol = bv[col];
#pragma unroll
    for (int j = 0; j < 8; ++j) {
        const int row = r0 + j + 8 * half;
        const float val = acc[j] + sc[row] * bcol;
        const int bb2 = row / 3, jj = row - 3 * bb2;
        C[(size_t)bb2 * (3 * DV_) + jj * DV_ + col] = val;
    }
}

// ---------------------------------------------------------------------------
// 5) Generic 16xN = 16xK @ (N x K)^T + bias (optional exact GELU), WMMA f32.
//    grid: N/64 blocks x 128 threads (4 waves, one 16x16 tile each).
// ---------------------------------------------------------------------------
__global__ void wmma_gemm16_kernel(const float* __restrict__ X,
                                   const float* __restrict__ W,
                                   const float* __restrict__ bias,
                                   float* __restrict__ Y,
                                   int K, int N, int doGelu) {
    const int wv = threadIdx.x >> 5, lane = threadIdx.x & 31;
    const int o0 = (blockIdx.x * 4 + wv) * 16;
    const int m = lane & 15, half = lane >> 4;
    const float* xp = X + (size_t)m * K + 2 * half;
    const float* wp = W + (size_t)(o0 + m) * K + 2 * half;
    v8f acc = {};
    for (int k = 0; k < K; k += 4) {
        const v2f a  = *(const v2f*)(xp + k);
        const v2f bb = *(const v2f*)(wp + k);
        acc = __builtin_amdgcn_wmma_f32_16x16x4_f32(false, a, false, bb, (short)0, acc, false, false);
    }
    const int col = o0 + m;
    const float bc = bias[col];
#pragma unroll
    for (int j = 0; j < 8; ++j) {
        const int row = j + 8 * half;
        float v = acc[j] + bc;
        if (doGelu) v = 0.5f * v * (1.0f + erff(v * 0.70710678118654752f));
        Y[(size_t)row * N + col] = v;
    }
}

// ---------------------------------------------------------------------------
extern "C" void kernel_launch(void* const* d_in, const int* in_sizes, int n_in,
                              void* d_out, int out_size, void* d_ws, size_t ws_size,
                              hipStream_t stream) {
    const float* F      = (const float*)d_in[0];
    const float* alpha  = (const float*)d_in[1];
    const float* Wv_w   = (const float*)d_in[2];
    const float* Wv_b   = (const float*)d_in[3];
    const float* fus_w1 = (const float*)d_in[4];
    const float* fus_b1 = (const float*)d_in[5];
    const float* fus_w2 = (const float*)d_in[6];
    const float* fus_b2 = (const float*)d_in[7];
    const int*   step   = (const int*)d_in[8];
    float* out = (float*)d_out;
    float* ws  = (float*)d_ws;

    topk_kernel<<<B_, 256, 0, stream>>>(alpha, ws);
    stream_reduce_kernel<<<dim3(SL, B_), 256, 0, stream>>>(F, alpha, ws);
    build_x_kernel<<<B_, 256, 0, stream>>>(F, ws);
    wmma_gemm_v_kernel<<<24, 128, 0, stream>>>(Wv_w, Wv_b, step, ws);
    // h = gelu(concat @ w1^T + b1): M=16, K=1536, N=1024
    wmma_gemm16_kernel<<<16, 128, 0, stream>>>(ws + OFF_CONCAT, fus_w1, fus_b1,
                                               ws + OFF_H, 3 * DV_, 1024, 1);
    // out = h @ w2^T + b2: M=16, K=1024, N=1024
    wmma_gemm16_kernel<<<16, 128, 0, stream>>>(ws + OFF_H, fus_w2, fus_b2,
                                               out, 1024, 1024, 0);
}